// VisionLanguageModel_33603824124095
// MI455X (gfx1250) — compile-verified
//
#include <hip/hip_runtime.h>

typedef __attribute__((ext_vector_type(2))) float v2f;
typedef __attribute__((ext_vector_type(8))) float v8f;

static constexpr int kB = 8, kS = 197, kD = 768, kG = 5, kP = 4;
static constexpr int kSD    = kS * kD;          // 151296
static constexpr int kNTile = kB * kG * kP;     // 160
static constexpr int kNCh   = 16;               // K-chunks per tile
static constexpr int kVec4  = kSD / 4;          // 37824 float4s per tile
static constexpr int kChV   = kVec4 / kNCh;     // 2364 float4s per chunk
static constexpr float kInvS = 1.0f / (float)kS;

// ---------------------------------------------------------------------------
// Wave32 sum-reduction via V_WMMA_F32_16X16X4_F32.
//   A[m,0] = v(lane m), A[m,2] = v(lane m+16), A[:,1]=A[:,3]=0 ; B = ones(4x16)
//   => D[m,n] = v(lane m) + v(lane m+16) for every column n.
// Lane n (n<16) holds D[0..7, n] in its 8 accumulator VGPRs; lane n+16 holds
// D[8..15, n]. Per-lane sum of those 8 + one cross-half shuffle = full sum.
// Call only with EXEC == all ones (WMMA requirement).
// ---------------------------------------------------------------------------
__device__ __forceinline__ float wave_reduce_sum_wmma(float v) {
  v2f a;    a[0] = v;    a[1] = 0.0f;
  v2f ones; ones[0] = 1.0f; ones[1] = 1.0f;
  v8f c = {};
  c = __builtin_amdgcn_wmma_f32_16x16x4_f32(
      /*neg_a=*/false, a, /*neg_b=*/false, ones,
      /*c_mod=*/(short)0, c, /*reuse_a=*/false, /*reuse_b=*/false);
  float h = ((c[0] + c[1]) + (c[2] + c[3])) + ((c[4] + c[5]) + (c[6] + c[7]));
  h += __shfl_xor(h, 16, 32);
  return h;  // full 32-lane sum, replicated to all lanes
}

// ---------------------------------------------------------------------------
// K1: o_vec[b,d] = mean over S of visual_embeds[b,s,d]
// ---------------------------------------------------------------------------
__global__ void __launch_bounds__(256) k_mean(const float* __restrict__ vis,
                                              float* __restrict__ o_vec) {
  int idx = blockIdx.x * 256 + threadIdx.x;          // 0 .. B*D-1 (6144)
  int b = idx / kD, d = idx % kD;
  const float* p = vis + (size_t)b * kSD + d;
  float s = 0.0f;
#pragma unroll 4
  for (int t = 0; t < kS; ++t) s += p[t * kD];       // coalesced over d
  o_vec[idx] = s * kInvS;
}

// ---------------------------------------------------------------------------
// K2: score_t[b,i] = || sent[gold[b,i]] - o_vec[b] ||^2 ; one wave per (b,i)
// ---------------------------------------------------------------------------
__global__ void __launch_bounds__(256) k_score_t(const float* __restrict__ sent,
                                                 const float* __restrict__ o_vec,
                                                 const int*   __restrict__ gold,
                                                 float*       __restrict__ score_t) {
  int wave = (blockIdx.x * 256 + threadIdx.x) >> 5;  // 0..39 == b*G+i
  int lane = threadIdx.x & 31;
  int b   = wave / kG;
  int gid = gold[wave];
  const float* t = sent  + (size_t)gid * kD;
  const float* o = o_vec + (size_t)b   * kD;
  float acc = 0.0f;
  for (int d = lane; d < kD; d += 32) {
    float df = t[d] - o[d];
    acc = fmaf(df, df, acc);
  }
  float tot = wave_reduce_sum_wmma(acc);
  if (lane == 0) score_t[wave] = tot;
}

// ---------------------------------------------------------------------------
// K3 (heavy, bandwidth-bound): per (tile=b,i,j ; chunk) partial of
//   Sum_{s,d} (R[img,s,d] - o[b,s,d])^2   via float4 loads + FMA.
// ---------------------------------------------------------------------------
__global__ void __launch_bounds__(256) k_dist(const float* __restrict__ vis,
                                              const float* __restrict__ img,
                                              const int*   __restrict__ gold,
                                              const int*   __restrict__ s2i,
                                              float*       __restrict__ partials) {
  int tile = blockIdx.x;                 // 0..159
  int ch   = blockIdx.y;                 // 0..15
  int tid  = threadIdx.x;
  int b = tile / (kG * kP);
  int rem = tile % (kG * kP);
  int i = rem / kP, j = rem % kP;
  int gid = gold[b * kG + i];
  int iid = s2i[gid * kP + j];

  const float4* Rp = (const float4*)(img + (size_t)iid * kSD);
  const float4* Op = (const float4*)(vis + (size_t)b   * kSD);

  int start = ch * kChV, end = start + kChV;
  float acc = 0.0f;
  for (int v = start + tid; v < end; v += 256) {
    __builtin_prefetch((const void*)(Rp + v + 512), 0, 0);  // global_prefetch_b8
    float4 r = Rp[v];
    float4 o = Op[v];
    float d0 = r.x - o.x, d1 = r.y - o.y, d2 = r.z - o.z, d3 = r.w - o.w;
    acc = fmaf(d0, d0, acc);
    acc = fmaf(d1, d1, acc);
    acc = fmaf(d2, d2, acc);
    acc = fmaf(d3, d3, acc);
  }

  // Reconverged here: full-EXEC WMMA wave reduction, then LDS fold of 8 waves.
  float w = wave_reduce_sum_wmma(acc);
  __shared__ float wsum[8];
  int wid = tid >> 5, lane = tid & 31;
  if (lane == 0) wsum[wid] = w;
  __syncthreads();
  if (tid < 32) {
    float x = (lane < 8) ? wsum[lane] : 0.0f;
    x += __shfl_xor(x, 1, 32);
    x += __shfl_xor(x, 2, 32);
    x += __shfl_xor(x, 4, 32);
    if (lane == 0) partials[tile * kNCh + ch] = x;
  }
}

// ---------------------------------------------------------------------------
// K4: fold partials -> dist ; score_r = min_j ; combine ; mean over G -> out[8]
// ---------------------------------------------------------------------------
__global__ void __launch_bounds__(256) k_final(const float* __restrict__ partials,
                                               const float* __restrict__ score_t,
                                               float*       __restrict__ out) {
  __shared__ float distl[kNTile];
  __shared__ float combl[kB * kG];
  int t = threadIdx.x;
  if (t < kNTile) {
    float s = 0.0f;
    for (int c = 0; c < kNCh; ++c) s += partials[t * kNCh + c];
    distl[t] = s;
  }
  __syncthreads();
  if (t < kB * kG) {
    float m = distl[t * kP];
    for (int j = 1; j < kP; ++j) m = fminf(m, distl[t * kP + j]);
    combl[t] = 0.5f * m + 0.5f * score_t[t];   // GAMMA_R = GAMMA_T = 0.5
  }
  __syncthreads();
  if (t < kB) {
    float s = 0.0f;
    for (int i = 0; i < kG; ++i) s += combl[t * kG + i];
    out[t] = s * (1.0f / (float)kG);
  }
}

// ---------------------------------------------------------------------------
extern "C" void kernel_launch(void* const* d_in, const int* in_sizes, int n_in,
                              void* d_out, int out_size, void* d_ws, size_t ws_size,
                              hipStream_t stream) {
  const float* vis  = (const float*)d_in[0];   // (B,S,D)
  const float* sent = (const float*)d_in[1];   // (N_SENT,D)
  const float* img  = (const float*)d_in[2];   // (N_IMG,S,D)
  const int*   gold = (const int*)d_in[3];     // (B,G)
  const int*   s2i  = (const int*)d_in[4];     // (N_SENT,P)
  float* out = (float*)d_out;                  // (B,)

  float* o_vec    = (float*)d_ws;              // B*D   = 6144 floats
  float* score_t  = o_vec + kB * kD;           // B*G   = 40 floats
  float* partials = score_t + kB * kG;         // 160*16 = 2560 floats

  k_mean   <<<dim3((kB * kD) / 256), dim3(256), 0, stream>>>(vis, o_vec);
  k_score_t<<<dim3((kB * kG) / 8),   dim3(256), 0, stream>>>(sent, o_vec, gold, score_t);
  k_dist   <<<dim3(kNTile, kNCh),    dim3(256), 0, stream>>>(vis, img, gold, s2i, partials);
  k_final  <<<dim3(1),               dim3(256), 0, stream>>>(partials, score_t, out);
}